// SAUSAGE_62148176773493
// MI455X (gfx1250) — compile-verified
//
#include <hip/hip_runtime.h>
#include <hip/hip_bf16.h>
#include <math.h>

typedef __attribute__((ext_vector_type(2))) float v2f;
typedef __attribute__((ext_vector_type(8))) float v8f;

#define UNITS 512
#define DIM   64
#define BATCH 2048

// Workspace layout (floats):
// [0*UNITS)  c      = q1 . q12
// [1*UNITS)  q2d    = q2 . q12
// [2*UNITS)  q2n2   = ||q2||^2
// [3*UNITS)  q12n2  = ||q12||^2
// [4*UNITS)  q12nm  = ||q12||
// [5*UNITS)  invnr2 = 1 / (DIM * r^2)
// [6*UNITS)  xn2[BATCH] = ||x_b||^2
__global__ __launch_bounds__(256)
void sausage_precompute(const float* __restrict__ x,
                        const float* __restrict__ q1,
                        const float* __restrict__ q2,
                        const float* __restrict__ r,
                        float* __restrict__ ws) {
    int gid = blockIdx.x * blockDim.x + threadIdx.x;
    if (gid < UNITS) {
        const int u = gid;
        float c = 0.f, q2d = 0.f, q2n = 0.f, q12n = 0.f;
        #pragma unroll 8
        for (int k = 0; k < DIM; ++k) {
            float a = q1[u * DIM + k];
            float b = q2[u * DIM + k];
            float d = b - a;
            c    = fmaf(a, d, c);
            q2d  = fmaf(b, d, q2d);
            q2n  = fmaf(b, b, q2n);
            q12n = fmaf(d, d, q12n);
        }
        ws[0 * UNITS + u] = c;
        ws[1 * UNITS + u] = q2d;
        ws[2 * UNITS + u] = q2n;
        ws[3 * UNITS + u] = q12n;
        ws[4 * UNITS + u] = sqrtf(q12n);
        float rr = r[u];
        ws[5 * UNITS + u] = 1.0f / ((float)DIM * rr * rr);
    }
    int xi = gid - UNITS;
    if (xi >= 0 && xi < BATCH) {
        float s = 0.f;
        #pragma unroll 8
        for (int k = 0; k < DIM; ++k) {
            float v = x[xi * DIM + k];
            s = fmaf(v, v, s);
        }
        ws[6 * UNITS + xi] = s;
    }
}

// One wave (32 lanes) computes a 16x16 output tile via V_WMMA_F32_16X16X4_F32.
// 4 waves per block -> 64 consecutive units per block (x-tile reuse from L2).
__global__ __launch_bounds__(128)
void sausage_wmma(const float* __restrict__ x,
                  const float* __restrict__ q1,
                  const float* __restrict__ q2,
                  const float* __restrict__ ws,
                  float* __restrict__ out) {
    const int lane = threadIdx.x & 31;
    const int wave = threadIdx.x >> 5;
    const int row  = lane & 15;   // M (for A) / N (for B) within the tile
    const int hi   = lane >> 4;   // K-pair select: 0 -> K {0,1}, 1 -> K {2,3}

    const int u0 = (blockIdx.x * 4 + wave) * 16;  // unit tile origin
    const int b0 = blockIdx.y * 16;               // batch tile origin

    // A 16x4 f32 layout: lane<16 holds A[row][4k+{0,1}], lane>=16 holds A[row][4k+{2,3}]
    // B 4x16 f32 layout mirrors it with N=lane -> identical addressing on q1/q2.
    const float* ax = x  + (size_t)(b0 + row) * DIM + 2 * hi;
    const float* p1 = q1 + (size_t)(u0 + row) * DIM + 2 * hi;
    const float* p2 = q2 + (size_t)(u0 + row) * DIM + 2 * hi;

    v8f acc1 = {};  // G1 = x . q12^T
    v8f acc2 = {};  // G2 = x . q2^T

    #pragma unroll
    for (int k0 = 0; k0 < DIM / 4; ++k0) {
        const int off = 4 * k0;
        v2f a   = *(const v2f*)(ax + off);
        v2f bq1 = *(const v2f*)(p1 + off);
        v2f bq2 = *(const v2f*)(p2 + off);
        v2f bqd = bq2 - bq1;  // q12 fragment, built in registers
        acc1 = __builtin_amdgcn_wmma_f32_16x16x4_f32(
            false, a, false, bqd, (short)0, acc1, false, false);
        acc2 = __builtin_amdgcn_wmma_f32_16x16x4_f32(
            false, a, false, bq2, (short)0, acc2, false, false);
    }

    // C/D layout: VGPR j -> row (b0 + j + 8*hi), col (u0 + (lane&15))
    const int u = u0 + row;
    const float c     = ws[0 * UNITS + u];
    const float q2d   = ws[1 * UNITS + u];
    const float q2n   = ws[2 * UNITS + u];
    const float q12n  = ws[3 * UNITS + u];
    const float q12nm = ws[4 * UNITS + u];
    const float inv   = ws[5 * UNITS + u];
    const float* __restrict__ xn2 = ws + 6 * UNITS;

    #pragma unroll
    for (int j = 0; j < 8; ++j) {
        const int b = b0 + j + 8 * hi;
        float G1 = acc1[j];
        float G2 = acc2[j];
        float s  = G1 - c;
        float as = fabsf(s);
        // l = 1 if s<0 ; |s| if |s|<||q12|| ; else 0
        float l  = (s < 0.f) ? 1.f : ((as < q12nm) ? as : 0.f);
        // ||x - center||^2 = ||x||^2 - 2 x.q2 + ||q2||^2 + 2 l (x.q12 - q2.q12) + l^2 ||q12||^2
        float dd = xn2[b] - 2.f * G2 + q2n + 2.f * l * (G1 - q2d) + l * l * q12n;
        out[(size_t)b * UNITS + u] = __expf(-dd * inv);
    }
}

extern "C" void kernel_launch(void* const* d_in, const int* in_sizes, int n_in,
                              void* d_out, int out_size, void* d_ws, size_t ws_size,
                              hipStream_t stream) {
    const float* x  = (const float*)d_in[0];  // (BATCH, DIM)
    const float* q1 = (const float*)d_in[1];  // (UNITS, DIM)
    const float* q2 = (const float*)d_in[2];  // (UNITS, DIM)
    const float* r  = (const float*)d_in[3];  // (UNITS,)
    float* out = (float*)d_out;               // (BATCH, UNITS)
    float* ws  = (float*)d_ws;                // 6*UNITS + BATCH floats = 20 KB

    const int pre_threads = UNITS + BATCH;
    sausage_precompute<<<(pre_threads + 255) / 256, 256, 0, stream>>>(x, q1, q2, r, ws);

    dim3 grid(UNITS / 64, BATCH / 16);  // (8, 128)
    sausage_wmma<<<grid, 128, 0, stream>>>(x, q1, q2, ws, out);
}